// ForbiddenSeqActivation_32358283608221
// MI455X (gfx1250) — compile-verified
//
#include <hip/hip_runtime.h>
#include <stdint.h>

// Problem geometry (reference: B=4096, L=8192, WINDOW=8, fp32 +/-1 values).
#define L_LEN 8192
#define WORDS_PER_ROW (L_LEN / 32)         // 256 packed uint32 words per row
#define VECS_PER_ROW  (WORDS_PER_ROW / 4)  // 64 uint4 per row

// ---------------------------------------------------------------------------
// Pass 1: pack sign bits. One lane per float; wave32 ballot packs 32
// consecutive elements' sign bits into one uint32. Fully coalesced b32 loads.
// Bit j of word w == sign(x[32*w + j]) (1 => +1, 0 => -1).
// ---------------------------------------------------------------------------
__global__ void pack_bits_kernel(const float* __restrict__ x,
                                 uint32_t* __restrict__ packed) {
    const size_t gid = (size_t)blockIdx.x * blockDim.x + threadIdx.x;
    const float v = x[gid];
    const uint32_t ball = __builtin_amdgcn_ballot_w32(v > 0.0f);
    if ((threadIdx.x & 31u) == 0u) {
        packed[gid >> 5] = ball;
    }
}

// ---------------------------------------------------------------------------
// Pass 2: per-row sequential FSM in the DIFF domain (branchless, VCC-free).
//   a_t = x_t ^ x_{t-1}            (input diffs, word-parallel)
//   c_t = a_t ^ m_{t-1}            (candidate output diff)
//   m_t = [ last 6 output diffs all == c_t ]   (the 8-window match test)
//       -> q = (dwin ^ (0-c)) & 0x3F ; m = (q-1) >> 31
//   d_t = c_t ^ m_t                (output diff)
//   y   = prefix-xor(d) ^ y_prev   (log-step reconstruction per word)
// State carried across words: dwin (last 6 d's), m_prev, y_prev, x_carry.
// ---------------------------------------------------------------------------
struct FsmState {
    uint32_t dwin;     // trailing output diffs, bit0 = newest (only low 6 used)
    uint32_t m_prev;   // match flag of previous step (0/1)
    uint32_t y_prev;   // previous output bit (0/1)
    uint32_t x_carry;  // previous INPUT bit (0/1)
};

__device__ __forceinline__ uint32_t fsm32d(uint32_t xw, FsmState& st) {
    const uint32_t aw = xw ^ ((xw << 1) | st.x_carry);   // input diffs
    uint32_t dwin = st.dwin;
    uint32_t m    = st.m_prev;
    uint32_t dword = 0;
#pragma unroll
    for (int i = 0; i < 32; ++i) {
        const uint32_t a = (aw >> i) & 1u;
        const uint32_t c = a ^ m;
        const uint32_t q = (dwin ^ (0u - c)) & 0x3Fu;    // 6-way equality test
        m = (q - 1u) >> 31;                              // 1 iff q == 0
        const uint32_t d = c ^ m;
        dwin = (dwin << 1) | d;                          // overflow masked in q
        dword |= d << i;
    }
    // Reconstruct outputs: y_i = y_prev ^ (d_0 ^ ... ^ d_i)
    uint32_t y = dword;
    y ^= y << 1;  y ^= y << 2;  y ^= y << 4;  y ^= y << 8;  y ^= y << 16;
    y ^= (0u - st.y_prev);
    st.dwin = dwin;
    st.m_prev = m;
    st.y_prev = y >> 31;
    st.x_carry = xw >> 31;
    return y;
}

__global__ void fsm_kernel(uint32_t* __restrict__ packed) {
    const int row = blockIdx.x * blockDim.x + threadIdx.x;   // 4096 rows
    uint4* __restrict__ p4 =
        (uint4*)(packed + (size_t)row * WORDS_PER_ROW);      // 1KB-aligned

    // --- Word 0: direct-form bootstrap (carry init = x[:, :7] passthrough) ---
    uint4 v = p4[0];
    const uint32_t w0 = v.x;
    uint32_t win = 0;
#pragma unroll
    for (int i = 0; i < 7; ++i) win = (win << 1) | ((w0 >> i) & 1u);
    uint32_t out0 = w0 & 0x7Fu;
    uint32_t mlast = 0;
#pragma unroll
    for (int i = 7; i < 32; ++i) {
        const uint32_t xb = (w0 >> i) & 1u;
        const uint32_t w8 = (win << 1) | xb;                 // win<=0x7F
        const uint32_t q  = (w8 ^ (0u - xb)) & 0xFFu;        // in {0,0xAA} iff match
        const uint32_t m  = (uint32_t)((q == 0u) || (q == 0xAAu));
        win = (w8 ^ m) & 0x7Fu;
        out0 |= (xb ^ m) << i;
        mlast = m;
    }
    v.x = out0;

    // Convert window state -> diff-domain state.
    FsmState st;
    st.dwin    = (win ^ (win >> 1)) & 0x3Fu;  // last 6 output diffs
    st.m_prev  = mlast;
    st.y_prev  = out0 >> 31;
    st.x_carry = w0 >> 31;                    // last INPUT bit of word 0

    v.y = fsm32d(v.y, st);
    v.z = fsm32d(v.z, st);
    v.w = fsm32d(v.w, st);
    p4[0] = v;

    // --- Remaining 63 vectors: 128 FSM bits per b128 load/store, in place ---
    for (int wi = 1; wi < VECS_PER_ROW; ++wi) {
        uint4 w = p4[wi];
        if (wi + 8 < VECS_PER_ROW) {
            __builtin_prefetch(p4 + wi + 8, 0, 0);   // global_prefetch_b8
        }
        w.x = fsm32d(w.x, st);
        w.y = fsm32d(w.y, st);
        w.z = fsm32d(w.z, st);
        w.w = fsm32d(w.w, st);
        p4[wi] = w;
    }
}

// ---------------------------------------------------------------------------
// Pass 3: expand packed result bits to float {0,1}.
// Each 256-thread block covers 64 words (2048 elements). The 64 words are
// staged into LDS via the CDNA5 async global->LDS copy (the LDS address is
// passed INTO the asm so the compiler sees the LDS object escape and keeps
// the ds_load on the consumer side), then every thread expands one byte into
// 8 floats written as two coalesced b128 stores.
// ---------------------------------------------------------------------------
__global__ void expand_kernel(const uint32_t* __restrict__ packed,
                              float* __restrict__ out) {
    __shared__ uint32_t lds[64];
    const uint32_t tid = threadIdx.x;                 // 256
    const size_t word_base = (size_t)blockIdx.x * 64;

    if (tid < 64u) {
        const uint32_t goff = (uint32_t)((word_base + tid) * 4u); // byte off (4MB buffer)
        // Generic pointer to a __shared__ object is {aperture_hi, lds_offset};
        // low 32 bits are exactly the LDS byte offset VDST needs.
        const uint32_t ldsoff = (uint32_t)(uintptr_t)(&lds[tid]);
        asm volatile("global_load_async_to_lds_b32 %0, %1, %2"
                     :: "v"(ldsoff), "v"(goff), "s"(packed)
                     : "memory");
    }
    asm volatile("s_wait_asynccnt 0" ::: "memory");
    __syncthreads();

    const uint32_t word = lds[tid >> 2];              // ds_load_b32 (broadcast x4)
    const uint32_t byte = (word >> (8u * (tid & 3u))) & 0xFFu;

    float4 lo, hi;
    lo.x = (float)( byte        & 1u);
    lo.y = (float)((byte >> 1u) & 1u);
    lo.z = (float)((byte >> 2u) & 1u);
    lo.w = (float)((byte >> 3u) & 1u);
    hi.x = (float)((byte >> 4u) & 1u);
    hi.y = (float)((byte >> 5u) & 1u);
    hi.z = (float)((byte >> 6u) & 1u);
    hi.w = (float)((byte >> 7u) & 1u);

    const size_t e = word_base * 32u + (size_t)tid * 8u;   // 32B-aligned
    float4* o = (float4*)(out + e);
    o[0] = lo;
    o[1] = hi;
}

// ---------------------------------------------------------------------------
// Launch: pack -> FSM (in place) -> expand.
// Workspace use: n/8 bytes (4 MB) at the start of d_ws.
// ---------------------------------------------------------------------------
extern "C" void kernel_launch(void* const* d_in, const int* in_sizes, int n_in,
                              void* d_out, int out_size, void* d_ws, size_t ws_size,
                              hipStream_t stream) {
    (void)n_in; (void)out_size; (void)ws_size;
    const float* x = (const float*)d_in[0];
    float* out = (float*)d_out;

    const size_t n    = (size_t)in_sizes[0];      // 4096 * 8192
    const int    rows = (int)(n / L_LEN);         // 4096
    uint32_t* packed  = (uint32_t*)d_ws;          // n/32 words = 4 MB

    // Pass 1: coalesced sign-bit packing (memory bound, ~128 MB read).
    pack_bits_kernel<<<(unsigned)(n / 256), 256, 0, stream>>>(x, packed);

    // Pass 2: one lane per row; 128 single-wave workgroups spread over WGPs.
    fsm_kernel<<<rows / 32, 32, 0, stream>>>(packed);

    // Pass 3: coalesced expansion (memory bound, ~128 MB write).
    const size_t words = n / 32;                  // 1,048,576
    expand_kernel<<<(unsigned)(words / 64), 256, 0, stream>>>(packed, out);
}